// ResGAT_48576080118129
// MI455X (gfx1250) — compile-verified
//
#include <hip/hip_runtime.h>
#include <hip/hip_bf16.h>

// ---------------------------------------------------------------------------
// ResGAT forward for MI455X (gfx1250, wave32).
//   - Dense 16-wide GEMMs via V_WMMA_F32_16X16X4_F32 (one wave = several 16x16
//     tiles; B fragments preloaded once per wave; guard-free fast path)
//   - GATv2 edge softmax via 3-pass atomic scheme (L2-resident)
// ---------------------------------------------------------------------------

typedef __attribute__((ext_vector_type(2))) float v2f;
typedef __attribute__((ext_vector_type(8))) float v8f;

__device__ __forceinline__ float leaky(float v, float s) { return v >= 0.f ? v : s * v; }

// Orderable-key mapping for float atomic max on unsigned.
__device__ __forceinline__ unsigned f2key(float f) {
    unsigned b = __float_as_uint(f);
    return (b & 0x80000000u) ? ~b : (b | 0x80000000u);
}
__device__ __forceinline__ float key2f(unsigned k) {
    unsigned b = (k & 0x80000000u) ? (k & 0x7FFFFFFFu) : ~k;
    return __uint_as_float(b);
}

// ---------------------------------------------------------------------------
// One 16x16 output tile: K/4 chained WMMA f32 16x16x4 ops.
// GUARD=false: no per-lane guards anywhere (EXEC untouched, immediate-offset
// stores). GUARD=true: clamped loads + per-row store guards (last tile only).
// WMMA f32 16x16x4 fragment layout (ISA 7.12.2):
//   A: lanes 0-15 hold K={0,1}, lanes 16-31 K={2,3}, M = lane&15
//   B: symmetric, N = lane&15
//   C/D: VGPR j -> M = j + (lane>=16 ? 8 : 0), N = lane&15
// ---------------------------------------------------------------------------
template <int K, int LDA, int FOUT, int ACT, bool GUARD>
__device__ __forceinline__ void gemm_tile(const float* __restrict__ A0,
                                          const float* __restrict__ A1,
                                          const v2f* bf, float bv,
                                          float* __restrict__ out,
                                          int nrows, int base,
                                          int mrow, int half, int col)
{
    constexpr int NKB = K / 4;
    int row  = base + mrow;
    int srow = GUARD ? (row < nrows ? row : nrows - 1) : row;

    v8f c = {};
#pragma unroll
    for (int kb = 0; kb < NKB; ++kb) {
        int k = kb * 4 + half * 2;
        const float* ap = (kb < 4) ? (A0 + (size_t)srow * LDA + k)
                                   : (A1 + (size_t)srow * LDA + (k - 16));
        v2f a;
        a.x = ap[0];
        a.y = ap[1];
        c = __builtin_amdgcn_wmma_f32_16x16x4_f32(false, a, false, bf[kb],
                                                  (short)0, c, false, false);
    }

    float* po = out + (size_t)(base + half * 8) * FOUT + col;
    if (FOUT == 16 || col < FOUT) {
#pragma unroll
        for (int j = 0; j < 8; ++j) {
            int r = base + j + half * 8;
            if (!GUARD || r < nrows) {
                float v = c[j] + bv;
                if (ACT) v = leaky(v, 0.01f);
                po[(size_t)j * FOUT] = v;   // constant immediate offset
            }
        }
    }
}

// GEMM: out[r,c] = act( sum_k A[r,k] * W[k,c] + bias[c] )
//   A columns 0..15 from A0, 16..31 from A1; both row stride LDA.
template <int K, int LDA, int FOUT, int ACT>
__global__ void gemm16_wmma(const float* __restrict__ A0,
                            const float* __restrict__ A1,
                            const float* __restrict__ W,
                            const float* __restrict__ bias,
                            float* __restrict__ out,
                            int nrows, int tpw)
{
    constexpr int NKB = K / 4;
    int wave = (blockIdx.x * blockDim.x + threadIdx.x) >> 5;
    int lane = threadIdx.x & 31;
    int mrow = lane & 15;
    int half = lane >> 4;
    int col  = mrow;

    // ---- Preload B fragments once per wave (branchless, EXEC untouched) ----
    v2f bf[NKB];
#pragma unroll
    for (int kb = 0; kb < NKB; ++kb) {
        int k = kb * 4 + half * 2;
        if constexpr (FOUT == 16) {
            bf[kb].x = W[(k + 0) * 16 + col];
            bf[kb].y = W[(k + 1) * 16 + col];
        } else {
            int   cc = col < FOUT ? col : 0;
            float mk = col < FOUT ? 1.f : 0.f;
            bf[kb].x = W[(k + 0) * FOUT + cc] * mk;
            bf[kb].y = W[(k + 1) * FOUT + cc] * mk;
        }
    }
    float bv = 0.f;
    if (bias) bv = bias[col < FOUT ? col : 0];

    int tile0 = wave * tpw;
    for (int t = 0; t < tpw; ++t) {
        int base = (tile0 + t) * 16;
        if (base >= nrows) return;               // wave-uniform
        if (base + 16 <= nrows) {                // wave-uniform fast path
            gemm_tile<K, LDA, FOUT, ACT, false>(A0, A1, bf, bv, out, nrows,
                                                base, mrow, half, col);
        } else {
            gemm_tile<K, LDA, FOUT, ACT, true>(A0, A1, bf, bv, out, nrows,
                                               base, mrow, half, col);
        }
    }
}

// ---------------------------------------------------------------------------
// GATv2 pass 1: e = dot(leaky(xl[src]+xr[dst], 0.2), att); segment max via
// atomic max on orderable keys. Edges id >= E are the N self-loops.
// ---------------------------------------------------------------------------
__global__ void gat_edge_score(const int* __restrict__ src, const int* __restrict__ dst,
                               int E, int n,
                               const float* __restrict__ xl, const float* __restrict__ xr,
                               const float* __restrict__ att,
                               float* __restrict__ ebuf, unsigned* __restrict__ emax)
{
    int id = blockIdx.x * blockDim.x + threadIdx.x;
    int tot = E + n;
    if (id >= tot) return;
    int s, d;
    if (id < E) { s = src[id]; d = dst[id]; } else { s = id - E; d = s; }

    const float4* pl = (const float4*)(xl + (size_t)s * 16);
    const float4* pr = (const float4*)(xr + (size_t)d * 16);
    const float4* pa = (const float4*)att;
    float e = 0.f;
#pragma unroll
    for (int q = 0; q < 4; ++q) {
        float4 a = pl[q], b = pr[q], w = pa[q];
        e += leaky(a.x + b.x, 0.2f) * w.x;
        e += leaky(a.y + b.y, 0.2f) * w.y;
        e += leaky(a.z + b.z, 0.2f) * w.z;
        e += leaky(a.w + b.w, 0.2f) * w.w;
    }
    ebuf[id] = e;
    atomicMax(emax + d, f2key(e));
}

// ---------------------------------------------------------------------------
// GATv2 pass 2: ex = exp(e - emax[dst]); accumulate denominator and
// un-normalized numerator sum(ex * xl[src]) into dst via float atomics.
// ---------------------------------------------------------------------------
__global__ void gat_edge_scatter(const int* __restrict__ src, const int* __restrict__ dst,
                                 int E, int n,
                                 const float* __restrict__ xl,
                                 const float* __restrict__ ebuf,
                                 const unsigned* __restrict__ emax,
                                 float* __restrict__ acc, float* __restrict__ denom)
{
    int id = blockIdx.x * blockDim.x + threadIdx.x;
    int tot = E + n;
    if (id >= tot) return;
    int s, d;
    if (id < E) { s = src[id]; d = dst[id]; } else { s = id - E; d = s; }

    float ex = __expf(ebuf[id] - key2f(emax[d]));
    atomicAdd(denom + d, ex);
    const float* pl = xl + (size_t)s * 16;
    float* pa = acc + (size_t)d * 16;
#pragma unroll
    for (int c = 0; c < 16; ++c) atomicAdd(pa + c, ex * pl[c]);
}

// GATv2 pass 3: out = acc / denom + bias   (one thread per node-channel)
__global__ void gat_finalize(const float* __restrict__ acc, const float* __restrict__ denom,
                             const float* __restrict__ bias, float* __restrict__ out, int n)
{
    int i = blockIdx.x * blockDim.x + threadIdx.x;
    int node = i >> 4;
    if (node >= n) return;
    out[i] = acc[i] / denom[node] + bias[i & 15];
}

// out = layernorm(a + b) * g + beta   (16-channel LN, one thread per node)
__global__ void ln_residual(const float* __restrict__ a, const float* __restrict__ b,
                            const float* __restrict__ g, const float* __restrict__ be,
                            float* __restrict__ out, int n)
{
    int i = blockIdx.x * blockDim.x + threadIdx.x;
    if (i >= n) return;
    const float4* pa = (const float4*)(a + (size_t)i * 16);
    const float4* pb = (const float4*)(b + (size_t)i * 16);
    float h[16];
    float m = 0.f;
#pragma unroll
    for (int q = 0; q < 4; ++q) {
        float4 x = pa[q], y = pb[q];
        h[4 * q + 0] = x.x + y.x;
        h[4 * q + 1] = x.y + y.y;
        h[4 * q + 2] = x.z + y.z;
        h[4 * q + 3] = x.w + y.w;
        m += h[4 * q + 0] + h[4 * q + 1] + h[4 * q + 2] + h[4 * q + 3];
    }
    m *= (1.f / 16.f);
    float v = 0.f;
#pragma unroll
    for (int c = 0; c < 16; ++c) { float dd = h[c] - m; v += dd * dd; }
    v *= (1.f / 16.f);
    float r = rsqrtf(v + 1e-5f);
    float* po = out + (size_t)i * 16;
#pragma unroll
    for (int c = 0; c < 16; ++c) po[c] = (h[c] - m) * r * g[c] + be[c];
}

// ---------------------------------------------------------------------------
struct GatP { const float *Wl, *Wr, *att, *bias; };
struct LayerP {
    GatP cl, cg;
    const float *mixW, *mixB, *gnG, *gnB;
    const float *m1W, *m1B, *m2W, *m2B, *fnG, *fnB;
};

extern "C" void kernel_launch(void* const* d_in, const int* in_sizes, int n_in,
                              void* d_out, int out_size, void* d_ws, size_t ws_size,
                              hipStream_t stream)
{
    (void)n_in; (void)out_size; (void)ws_size;

    const float* x   = (const float*)d_in[0];
    const int*   ei  = (const int*)d_in[1];
    const int*   gei = (const int*)d_in[2];
    const int N  = in_sizes[0] / 32;
    const int EL = in_sizes[1] / 2;
    const int EG = in_sizes[2] / 2;

    // Params flattened in setup_inputs() insertion order.
    int pi = 3;
    auto P = [&]() { return (const float*)d_in[pi++]; };
    const float *We1 = P(), *be1 = P(), *We2 = P(), *be2 = P();
    LayerP L[5];
    for (int l = 0; l < 5; ++l) {
        L[l].cl  = { P(), P(), P(), P() };   // conv_l: Wl, Wr, att, bias
        L[l].cg  = { P(), P(), P(), P() };   // conv_g
        L[l].mixW = P(); L[l].mixB = P();
        L[l].gnG  = P(); L[l].gnB  = P();
        L[l].m1W  = P(); L[l].m1B  = P();
        L[l].m2W  = P(); L[l].m2B  = P();
        L[l].fnG  = P(); L[l].fnB  = P();
    }
    const float *Wh1 = P(), *bh1 = P(), *Wh2 = P(), *bh2 = P(), *Wh3 = P(), *bh3 = P();

    // Workspace layout (floats). ~66 MB total.
    const size_t NB = (size_t)N * 16;
    float* ws  = (float*)d_ws;
    float* h   = ws;
    float* hn  = h  + NB;
    float* t0  = hn + NB;
    float* xl  = t0 + NB;
    float* xr  = xl + NB;
    float* x1  = xr + NB;
    float* x2  = x1 + NB;
    float* acc = x2 + NB;                       // NB floats
    float* denom = acc + NB;                    // N floats   } contiguous with acc
    unsigned* emax = (unsigned*)(denom + N);    // N keys     } for a single memset
    float* ebuf = (float*)(emax + N);           // max(EL,EG)+N floats

    const dim3 blk(256);
    constexpr int TPW = 4;                      // row tiles per wave

    auto launch_gemm = [&](auto kern, const float* A0, const float* A1,
                           const float* W, const float* bias, float* out, int nrows) {
        int tiles = (nrows + 15) / 16;
        int waves = (tiles + TPW - 1) / TPW;
        int thr   = waves * 32;
        kern<<<dim3((thr + 255) / 256), blk, 0, stream>>>(
            A0, A1, W, bias, out, nrows, TPW);
    };

    auto conv = [&](const float* xin, const int* eidx, int E, const GatP& g, float* outb) {
        launch_gemm(gemm16_wmma<16, 16, 16, 0>, xin, xin, g.Wl, nullptr, xl, N);
        launch_gemm(gemm16_wmma<16, 16, 16, 0>, xin, xin, g.Wr, nullptr, xr, N);
        hipMemsetAsync(acc, 0, (NB + 2 * (size_t)N) * sizeof(float), stream);
        int tot = E + N;
        gat_edge_score<<<dim3((tot + 255) / 256), blk, 0, stream>>>(
            eidx, eidx + E, E, N, xl, xr, g.att, ebuf, emax);
        gat_edge_scatter<<<dim3((tot + 255) / 256), blk, 0, stream>>>(
            eidx, eidx + E, E, N, xl, ebuf, emax, acc, denom);
        gat_finalize<<<dim3((unsigned)((NB + 255) / 256)), blk, 0, stream>>>(
            acc, denom, g.bias, outb, N);
    };

    // Embed MLP: leaky(x @ We1 + be1) @ We2 + be2   (x is N x 32, stride 32)
    launch_gemm(gemm16_wmma<32, 32, 16, 1>, x, x + 16, We1, be1, t0, N);
    launch_gemm(gemm16_wmma<16, 16, 16, 0>, t0, t0, We2, be2, h, N);

    for (int l = 0; l < 5; ++l) {
        conv(h, ei,  EL, L[l].cl, x1);
        conv(h, gei, EG, L[l].cg, x2);
        // mix_fc on concat([x1, x2])   (both stride 16)
        launch_gemm(gemm16_wmma<32, 16, 16, 0>, x1, x2, L[l].mixW, L[l].mixB, t0, N);
        ln_residual<<<dim3((N + 255) / 256), blk, 0, stream>>>(
            t0, h, L[l].gnG, L[l].gnB, hn, N);
        // fc_mlp + residual LN
        launch_gemm(gemm16_wmma<16, 16, 16, 1>, hn, hn, L[l].m1W, L[l].m1B, t0, N);
        launch_gemm(gemm16_wmma<16, 16, 16, 0>, t0, t0, L[l].m2W, L[l].m2B, x1, N);
        ln_residual<<<dim3((N + 255) / 256), blk, 0, stream>>>(
            x1, hn, L[l].fnG, L[l].fnB, h, N);
    }

    // Head: leaky(leaky(h@W1+b1)@W2+b2)@W3+b3  -> [N, 8]
    launch_gemm(gemm16_wmma<16, 16, 16, 1>, h,  h,  Wh1, bh1, t0, N);
    launch_gemm(gemm16_wmma<16, 16, 16, 1>, t0, t0, Wh2, bh2, x1, N);
    launch_gemm(gemm16_wmma<16, 16, 8, 0>,  x1, x1, Wh3, bh3, (float*)d_out, N);
}